// Net_46737834115140
// MI455X (gfx1250) — compile-verified
//
#include <hip/hip_runtime.h>
#include <hip/hip_bf16.h>

// ---------------- problem constants ----------------
constexpr int B_   = 2;
constexpr int L_   = 16384;
constexpr int C_   = 32;
constexpr int DIN  = 64;
constexpr int N_   = 16;
constexpr int HW_  = 128;     // image side (128x128 = L)
constexpr int LC   = 256;     // scan chunk length
constexpr int NCH  = L_ / LC; // 64 chunks

// workspace layout (float offsets)
constexpr size_t OFF_X  = 0;                              // x_raw   (B,L,64)
constexpr size_t OFF_XP = OFF_X  + (size_t)B_*L_*DIN;     // xp_raw  (B,L,64)
constexpr size_t OFF_Z  = OFF_XP + (size_t)B_*L_*DIN;     // z       (B,L,64)
constexpr size_t OFF_U  = OFF_Z  + (size_t)B_*L_*DIN;     // u=silu(conv(x)) (B,L,64)
constexpr size_t OFF_DT = OFF_U  + (size_t)B_*L_*DIN;     // dt      (B,L,64)
constexpr size_t OFF_BM = OFF_DT + (size_t)B_*L_*DIN;     // Bm      (B,L,16)
constexpr size_t OFF_CM = OFF_BM + (size_t)B_*L_*N_;      // Cm      (B,L,16)
constexpr size_t OFF_Y  = OFF_CM + (size_t)B_*L_*N_;      // y       (B,L,64)
constexpr size_t OFF_G  = OFF_Y  + (size_t)B_*L_*DIN;     // g       (B,L,32)
constexpr size_t OFF_AP = OFF_G  + (size_t)B_*L_*C_;      // chunk aProd (B,64,NCH,16)
constexpr size_t OFF_HE = OFF_AP + (size_t)B_*DIN*NCH*N_; // chunk hEnd
constexpr size_t OFF_HS = OFF_HE + (size_t)B_*DIN*NCH*N_; // chunk hStart

typedef __attribute__((ext_vector_type(2))) float v2f;
typedef __attribute__((ext_vector_type(8))) float v8f;

// fp32 WMMA:  D(16x16) = A(16x4) * B(4x16) + C  --- exact fp32 semantics
__device__ __forceinline__ v8f wmma4(v2f a, v2f b, v8f c) {
  return __builtin_amdgcn_wmma_f32_16x16x4_f32(
      /*neg_a=*/false, a, /*neg_b=*/false, b,
      /*c_mod=*/(short)0, c, /*reuse_a=*/false, /*reuse_b=*/false);
}

__device__ __forceinline__ float fast_silu(float x) {
  return x / (1.f + __expf(-x));
}

// ============================================================
// K1: resi add + double LN (ms and pan paths) + in_proj / in_proj_pan (WMMA)
//   256 threads, 128 tokens per block.  grid = B*L/128
// ============================================================
__global__ __launch_bounds__(256)
void k1_ln_inproj(const float* __restrict__ ms, const float* __restrict__ msresi,
                  const float* __restrict__ pan,
                  const float* __restrict__ ln1w, const float* __restrict__ ln1b,
                  const float* __restrict__ ln2w, const float* __restrict__ ln2b,
                  const float* __restrict__ Win,  const float* __restrict__ Winp,
                  float* __restrict__ out_resi,
                  float* __restrict__ xr, float* __restrict__ zr,
                  float* __restrict__ xpr)
{
  __shared__ float s_msn[128][33];
  __shared__ float s_pan[128][33];

  const int bid   = blockIdx.x;
  const int b     = bid / (L_ / 128);
  const int tbase = (bid % (L_ / 128)) * 128;
  const int tid   = threadIdx.x;

  // ---- LN phase: threads 0..127 -> ms stream, 128..255 -> pan stream ----
  {
    const int tok  = tid & 127;
    const long base = ((long)b * L_ + tbase + tok) * C_;
    float v[32];
    float s = 0.f;
    if (tid < 128) {
#pragma unroll
      for (int c = 0; c < 32; ++c) {
        float r = ms[base + c] + msresi[base + c];
        out_resi[base + c] = r;          // second tuple output
        v[c] = r; s += r;
      }
    } else {
#pragma unroll
      for (int c = 0; c < 32; ++c) { float r = pan[base + c]; v[c] = r; s += r; }
    }
    const float* w  = (tid < 128) ? ln1w : ln2w;
    const float* bb = (tid < 128) ? ln1b : ln2b;
#pragma unroll
    for (int pass = 0; pass < 2; ++pass) {       // reference applies LN twice
      float mu = s * (1.f / 32.f);
      float var = 0.f;
#pragma unroll
      for (int c = 0; c < 32; ++c) { float d = v[c] - mu; var += d * d; }
      var *= (1.f / 32.f);
      float inv = rsqrtf(var + 1e-5f);
      s = 0.f;
#pragma unroll
      for (int c = 0; c < 32; ++c) { v[c] = (v[c] - mu) * inv * w[c] + bb[c]; s += v[c]; }
    }
    if (tid < 128) {
#pragma unroll
      for (int c = 0; c < 32; ++c) s_msn[tok][c] = v[c];
    } else {
#pragma unroll
      for (int c = 0; c < 32; ++c) s_pan[tok][c] = v[c];
    }
  }
  __syncthreads();

  // ---- WMMA phase: 8 waves x 16 tokens ----
  const int wave = tid >> 5;
  const int lane = tid & 31;
  const int m    = lane & 15;
  const int kh   = (lane >> 4) * 2;         // K sub-offset per ISA A/B layout
  const int trow = wave * 16;

  v2f aM[8], aP[8];
#pragma unroll
  for (int k = 0; k < 8; ++k) {             // K=32 -> 8 steps of 4
    int kk = k * 4 + kh;
    v2f t0; t0.x = s_msn[trow + m][kk]; t0.y = s_msn[trow + m][kk + 1]; aM[k] = t0;
    v2f t1; t1.x = s_pan[trow + m][kk]; t1.y = s_pan[trow + m][kk + 1]; aP[k] = t1;
  }

  // xz = msn @ Win^T : 8 N-tiles (cols 0..63 -> x, 64..127 -> z)
#pragma unroll
  for (int nt = 0; nt < 8; ++nt) {
    v8f acc = {};
#pragma unroll
    for (int k = 0; k < 8; ++k) {
      int row = nt * 16 + m;
      int col = k * 4 + kh;
      v2f bf; bf.x = Win[row * 32 + col]; bf.y = Win[row * 32 + col + 1];
      acc = wmma4(aM[k], bf, acc);
    }
    float* dst = (nt < 4) ? xr : zr;
    int dcol = (nt < 4) ? nt * 16 + m : (nt - 4) * 16 + m;
#pragma unroll
    for (int r = 0; r < 8; ++r) {
      int t = tbase + trow + r + ((lane >> 4) ? 8 : 0);
      dst[((long)b * L_ + t) * DIN + dcol] = acc[r];
    }
  }

  // xp = pann @ Winp^T : 4 N-tiles
#pragma unroll
  for (int nt = 0; nt < 4; ++nt) {
    v8f acc = {};
#pragma unroll
    for (int k = 0; k < 8; ++k) {
      int row = nt * 16 + m;
      int col = k * 4 + kh;
      v2f bf; bf.x = Winp[row * 32 + col]; bf.y = Winp[row * 32 + col + 1];
      acc = wmma4(aP[k], bf, acc);
    }
#pragma unroll
    for (int r = 0; r < 8; ++r) {
      int t = tbase + trow + r + ((lane >> 4) ? 8 : 0);
      xpr[((long)b * L_ + t) * DIN + nt * 16 + m] = acc[r];
    }
  }
}

// ============================================================
// K2: causal conv1d(K=4)+silu for x and xp; x_proj (WMMA) -> dtr/Bm/Cm; dt_proj+softplus
//   128 threads (4 waves), 64 tokens per block + 3-token halo.  grid = B*L/64
// ============================================================
__global__ __launch_bounds__(128)
void k2_conv_xproj(const float* __restrict__ xr, const float* __restrict__ xpr,
                   const float* __restrict__ cw,  const float* __restrict__ cb,
                   const float* __restrict__ cpw, const float* __restrict__ cpb,
                   const float* __restrict__ Wxp,
                   const float* __restrict__ Wdt, const float* __restrict__ bdt,
                   float* __restrict__ u, float* __restrict__ dtv,
                   float* __restrict__ BmO, float* __restrict__ CmO)
{
  __shared__ float s_x[67 * 64];
  __shared__ float s_xp[67 * 64];
  __shared__ float s_xc[64][65];
  __shared__ float s_dtr[64][2];

  const int bid   = blockIdx.x;
  const int b     = bid / (L_ / 64);
  const int tbase = (bid % (L_ / 64)) * 64;
  const int tid   = threadIdx.x;

  // load tiles with left halo of 3 (zeros before t=0)
  for (int i = tid; i < 67 * 64; i += 128) {
    int t = tbase - 3 + (i >> 6);
    int d = i & 63;
    float vx = 0.f, vp = 0.f;
    if (t >= 0) {
      long idx = ((long)b * L_ + t) * DIN + d;
      vx = xr[idx]; vp = xpr[idx];
    }
    s_x[i] = vx; s_xp[i] = vp;
  }
  __syncthreads();

  // depthwise causal conv1d + silu
  for (int i = tid; i < 64 * 64; i += 128) {
    int t = i >> 6, d = i & 63;
    float ax = cb[d], ap = cpb[d];
#pragma unroll
    for (int j = 0; j < 4; ++j) {
      ax += cw[d * 4 + j]  * s_x[(t + j) * 64 + d];
      ap += cpw[d * 4 + j] * s_xp[(t + j) * 64 + d];
    }
    ax = fast_silu(ax);
    ap = fast_silu(ap);
    u[((long)b * L_ + tbase + t) * DIN + d] = ax;
    s_xc[t][d] = ap;
  }
  __syncthreads();

  // dbl = xp_conv @ Wxp^T  (34 cols -> 3 N-tiles, guarded)
  const int wave = tid >> 5;
  const int lane = tid & 31;
  const int m    = lane & 15;
  const int kh   = (lane >> 4) * 2;
  const int trow = wave * 16;

  v2f a[16];
#pragma unroll
  for (int k = 0; k < 16; ++k) {    // K=64 -> 16 steps
    int kk = k * 4 + kh;
    v2f t; t.x = s_xc[trow + m][kk]; t.y = s_xc[trow + m][kk + 1]; a[k] = t;
  }
#pragma unroll
  for (int nt = 0; nt < 3; ++nt) {
    v8f acc = {};
#pragma unroll
    for (int k = 0; k < 16; ++k) {
      int row = nt * 16 + m; if (row > 33) row = 33;   // clamp (results discarded)
      int col = k * 4 + kh;
      v2f bf; bf.x = Wxp[row * 64 + col]; bf.y = Wxp[row * 64 + col + 1];
      acc = wmma4(a[k], bf, acc);
    }
#pragma unroll
    for (int r = 0; r < 8; ++r) {
      int tl = trow + r + ((lane >> 4) ? 8 : 0);
      long t = (long)b * L_ + tbase + tl;
      int col = nt * 16 + m;
      if (col < 2)        s_dtr[tl][col] = acc[r];
      else if (col < 18)  BmO[t * N_ + col - 2]  = acc[r];
      else if (col < 34)  CmO[t * N_ + col - 18] = acc[r];
    }
  }
  __syncthreads();

  // dt = softplus(dtr @ Wdt^T + bdt)
  for (int i = tid; i < 64 * 64; i += 128) {
    int t = i >> 6, d = i & 63;
    float acc = bdt[d] + s_dtr[t][0] * Wdt[d * 2 + 0] + s_dtr[t][1] * Wdt[d * 2 + 1];
    float sp = (acc > 20.f) ? acc : __logf(1.f + __expf(acc));
    dtv[((long)b * L_ + tbase + t) * DIN + d] = sp;
  }
}

// ============================================================
// K3: chunked scan pass A — per (b,d,chunk) compute prod(dA) and local h_end (h0=0)
//   16 lanes per unit (lane = state n), 2 units per wave.  grid = 512 x 256
// ============================================================
__global__ __launch_bounds__(256)
void k3_scanA(const float* __restrict__ dtv, const float* __restrict__ u,
              const float* __restrict__ Bm,  const float* __restrict__ Alog,
              float* __restrict__ aprod, float* __restrict__ hend)
{
  const int tid  = threadIdx.x;
  const int lane = tid & 31;
  const int waveg = blockIdx.x * 8 + (tid >> 5);
  const int unit  = waveg * 2 + (lane >> 4);
  const int n = lane & 15;
  const int b  = unit / (DIN * NCH);
  const int rem = unit % (DIN * NCH);
  const int d  = rem / NCH;
  const int ch = rem % NCH;

  const float Adn = -__expf(Alog[d * N_ + n]);
  const long base  = ((long)b * L_ + ch * LC) * DIN + d;
  const long baseB = ((long)b * L_ + ch * LC) * N_ + n;

  float h = 0.f, ap = 1.f;
  for (int l = 0; l < LC; ++l) {
    if ((l & 15) == 0) {
      __builtin_prefetch(dtv + base + (long)(l + 32) * DIN, 0, 0);
      __builtin_prefetch(u   + base + (long)(l + 32) * DIN, 0, 0);
    }
    float dt = dtv[base + (long)l * DIN];
    float uv = u  [base + (long)l * DIN];
    float bv = Bm [baseB + (long)l * N_];
    float da = __expf(dt * Adn);
    h  = fmaf(da, h, dt * uv * bv);
    ap *= da;
  }
  const long o = ((long)(b * DIN + d) * NCH + ch) * N_ + n;
  aprod[o] = ap;
  hend[o]  = h;
}

// ============================================================
// K4: chunk-carry composition — serial over 64 chunks per (b,d,n).  2048 threads
// ============================================================
__global__ __launch_bounds__(256)
void k4_scanB(const float* __restrict__ aprod, const float* __restrict__ hend,
              float* __restrict__ hstart)
{
  const int idx = blockIdx.x * 256 + threadIdx.x;
  if (idx >= B_ * DIN * N_) return;
  const int n  = idx & 15;
  const int bd = idx >> 4;
  float h = 0.f;
  for (int ch = 0; ch < NCH; ++ch) {
    const long o = ((long)bd * NCH + ch) * N_ + n;
    hstart[o] = h;
    h = aprod[o] * h + hend[o];
  }
}

// ============================================================
// K5: scan pass C — replay with true h_start; y[t,d] = sum_n h*Cm + u*D
//   cross-lane reduce over 16 states via wave32 shfl_xor butterflies
// ============================================================
__global__ __launch_bounds__(256)
void k5_scanC(const float* __restrict__ dtv, const float* __restrict__ u,
              const float* __restrict__ Bm,  const float* __restrict__ Cm,
              const float* __restrict__ Alog, const float* __restrict__ Dp,
              const float* __restrict__ hstart, float* __restrict__ y)
{
  const int tid  = threadIdx.x;
  const int lane = tid & 31;
  const int waveg = blockIdx.x * 8 + (tid >> 5);
  const int unit  = waveg * 2 + (lane >> 4);
  const int n = lane & 15;
  const int b  = unit / (DIN * NCH);
  const int rem = unit % (DIN * NCH);
  const int d  = rem / NCH;
  const int ch = rem % NCH;

  const float Adn = -__expf(Alog[d * N_ + n]);
  const float Dd  = Dp[d];
  const long base  = ((long)b * L_ + ch * LC) * DIN + d;
  const long baseB = ((long)b * L_ + ch * LC) * N_ + n;

  float h = hstart[((long)(b * DIN + d) * NCH + ch) * N_ + n];
  for (int l = 0; l < LC; ++l) {
    if ((l & 15) == 0) {
      __builtin_prefetch(dtv + base + (long)(l + 32) * DIN, 0, 0);
      __builtin_prefetch(Bm  + baseB + (long)(l + 32) * N_, 0, 0);
    }
    float dt = dtv[base + (long)l * DIN];
    float uv = u  [base + (long)l * DIN];
    float bv = Bm [baseB + (long)l * N_];
    float cv = Cm [baseB + (long)l * N_];
    float da = __expf(dt * Adn);
    h = fmaf(da, h, dt * uv * bv);
    float contrib = h * cv;
    // reduce across 16 aligned lanes (masks < 16 stay inside each state group)
    contrib += __shfl_xor(contrib, 8);
    contrib += __shfl_xor(contrib, 4);
    contrib += __shfl_xor(contrib, 2);
    contrib += __shfl_xor(contrib, 1);
    if (n == 0) y[base + (long)l * DIN] = contrib + uv * Dd;
  }
}

// ============================================================
// K6: g = (y * silu(z)) @ Wout^T  (64->32) via WMMA.  grid = B*L/128 x 256
// ============================================================
__global__ __launch_bounds__(256)
void k6_outproj(const float* __restrict__ y, const float* __restrict__ z,
                const float* __restrict__ Wout, float* __restrict__ g)
{
  const int tid  = threadIdx.x;
  const int wave = tid >> 5;
  const int lane = tid & 31;
  const int m    = lane & 15;
  const int kh   = (lane >> 4) * 2;
  const long tok0 = (long)blockIdx.x * 128 + wave * 16;   // flat (b*L + t)

  v2f a[16];
#pragma unroll
  for (int k = 0; k < 16; ++k) {
    int kk = k * 4 + kh;
    long i0 = (tok0 + m) * DIN + kk;
    float y0 = y[i0], y1 = y[i0 + 1];
    float z0 = z[i0], z1 = z[i0 + 1];
    v2f t; t.x = y0 * fast_silu(z0); t.y = y1 * fast_silu(z1); a[k] = t;
  }
#pragma unroll
  for (int nt = 0; nt < 2; ++nt) {
    v8f acc = {};
#pragma unroll
    for (int k = 0; k < 16; ++k) {
      int row = nt * 16 + m;
      int col = k * 4 + kh;
      v2f bf; bf.x = Wout[row * 64 + col]; bf.y = Wout[row * 64 + col + 1];
      acc = wmma4(a[k], bf, acc);
    }
#pragma unroll
    for (int r = 0; r < 8; ++r) {
      long t = tok0 + r + ((lane >> 4) ? 8 : 0);
      g[t * C_ + nt * 16 + m] = acc[r];
    }
  }
}

// ============================================================
// K7: 3x3 depthwise conv (SAME, zero pad) + bias + residual -> out
// ============================================================
__global__ __launch_bounds__(256)
void k7_dwconv(const float* __restrict__ g, const float* __restrict__ w,
               const float* __restrict__ bias, float* __restrict__ out)
{
  const long idx = (long)blockIdx.x * 256 + threadIdx.x;   // over B*128*128*32
  const int c  = (int)(idx & 31);
  const long s = idx >> 5;
  const int wx = (int)(s & 127);
  const long s2 = s >> 7;
  const int hy = (int)(s2 & 127);
  const int b  = (int)(s2 >> 7);
  float acc = bias[c];
#pragma unroll
  for (int dh = -1; dh <= 1; ++dh) {
    int nh = hy + dh; if (nh < 0 || nh > 127) continue;
#pragma unroll
    for (int dw = -1; dw <= 1; ++dw) {
      int nw = wx + dw; if (nw < 0 || nw > 127) continue;
      acc += w[c * 9 + (dh + 1) * 3 + (dw + 1)] *
             g[(((long)b * L_) + (long)nh * HW_ + nw) * C_ + c];
    }
  }
  out[idx] = acc + g[idx];
}

// ============================================================
extern "C" void kernel_launch(void* const* d_in, const int* in_sizes, int n_in,
                              void* d_out, int out_size, void* d_ws, size_t ws_size,
                              hipStream_t stream)
{
  const float* ms   = (const float*)d_in[0];
  const float* resi = (const float*)d_in[1];
  const float* pan  = (const float*)d_in[2];
  const float* ln1w = (const float*)d_in[3];
  const float* ln1b = (const float*)d_in[4];
  const float* ln2w = (const float*)d_in[5];
  const float* ln2b = (const float*)d_in[6];
  const float* Win  = (const float*)d_in[7];   // (128,32)
  const float* Winp = (const float*)d_in[8];   // (64,32)
  const float* cw   = (const float*)d_in[9];   // (64,4)
  const float* cb   = (const float*)d_in[10];
  const float* cpw  = (const float*)d_in[11];
  const float* cpb  = (const float*)d_in[12];
  const float* Wxp  = (const float*)d_in[13];  // (34,64)
  const float* Wdt  = (const float*)d_in[14];  // (64,2)
  const float* bdt  = (const float*)d_in[15];
  const float* Alog = (const float*)d_in[16];  // (64,16)
  const float* Dp   = (const float*)d_in[17];
  const float* Wout = (const float*)d_in[18];  // (32,64)
  const float* dww  = (const float*)d_in[19];  // (32,1,3,3)
  const float* dwb  = (const float*)d_in[20];

  float* out     = (float*)d_out;
  float* outResi = out + (size_t)B_ * L_ * C_;  // tuple second output

  float* ws = (float*)d_ws;
  float* xr  = ws + OFF_X;
  float* xpr = ws + OFF_XP;
  float* zr  = ws + OFF_Z;
  float* uu  = ws + OFF_U;
  float* dtv = ws + OFF_DT;
  float* Bmw = ws + OFF_BM;
  float* Cmw = ws + OFF_CM;
  float* yw  = ws + OFF_Y;
  float* gw  = ws + OFF_G;
  float* ap  = ws + OFF_AP;
  float* he  = ws + OFF_HE;
  float* hs  = ws + OFF_HS;

  k1_ln_inproj<<<B_ * L_ / 128, 256, 0, stream>>>(ms, resi, pan, ln1w, ln1b, ln2w, ln2b,
                                                  Win, Winp, outResi, xr, zr, xpr);
  k2_conv_xproj<<<B_ * L_ / 64, 128, 0, stream>>>(xr, xpr, cw, cb, cpw, cpb,
                                                  Wxp, Wdt, bdt, uu, dtv, Bmw, Cmw);
  k3_scanA<<<(B_ * DIN * NCH) / 16, 256, 0, stream>>>(dtv, uu, Bmw, Alog, ap, he);
  k4_scanB<<<(B_ * DIN * N_ + 255) / 256, 256, 0, stream>>>(ap, he, hs);
  k5_scanC<<<(B_ * DIN * NCH) / 16, 256, 0, stream>>>(dtv, uu, Bmw, Cmw, Alog, Dp, hs, yw);
  k6_outproj<<<B_ * L_ / 128, 256, 0, stream>>>(yw, zr, Wout, gw);
  k7_dwconv<<<(B_ * L_ * C_) / 256, 256, 0, stream>>>(gw, dww, dwb, out);
}